// ModulatedDeformConv_14516989461140
// MI455X (gfx1250) — compile-verified
//
#include <hip/hip_runtime.h>

typedef float v2f __attribute__((ext_vector_type(2)));
typedef float v8f __attribute__((ext_vector_type(8)));

#define HH 64
#define WW 64
#define CC 64
#define COUT 64
#define KK 9
#define CK 576          // C*K
#define MSTRIDE 72      // LDS row stride in floats; 2*72 % 64 == 16 -> bank-split halves

__global__ __launch_bounds__(256) void dcnv2_wmma_kernel(
    const float* __restrict__ input,    // (B, C, H, W)
    const float* __restrict__ offset,   // (B, 2K, H, W)
    const float* __restrict__ mask,     // (B, K, H, W)
    const float* __restrict__ weight,   // (Cout, C, KH, KW) == (Cout, CK)
    const float* __restrict__ bias,     // (Cout)
    float* __restrict__ out)            // (B, Cout, H, W)
{
    extern __shared__ float M[];        // [CK][MSTRIDE] modulated samples

    const int tid  = threadIdx.x;
    const int blk  = blockIdx.x;
    const int b    = blk >> 6;          // 64 row-tiles per image
    const int tile = blk & 63;          // output row ho
    const int ho   = tile;
    const int pixg0 = tile * 64;        // global pixel base of this tile

    // ---------------- Phase A: build M[ck][pix] in LDS ----------------
    for (int idx = tid; idx < CK; idx += 256) {
        const int k   = idx >> 6;       // 0..8
        const int pix = idx & 63;       // 0..63
        const int wo  = pix;

        const float offy = offset[(((b * 18) + 2 * k)     * HH + ho) * WW + wo];
        const float offx = offset[(((b * 18) + 2 * k + 1) * HH + ho) * WW + wo];
        const float mm   = mask  [(((b *  9) +     k)     * HH + ho) * WW + wo];

        const float py = offy + (float)(ho - 1 + (k / 3));
        const float px = offx + (float)(wo - 1 + (k % 3));

        const float y0f = floorf(py), x0f = floorf(px);
        const float wy  = py - y0f,   wx  = px - x0f;
        const int   y0  = (int)y0f,   x0  = (int)x0f;
        const int   y1  = y0 + 1,     x1  = x0 + 1;

        const bool vy0 = (y0 >= 0) && (y0 < HH);
        const bool vy1 = (y1 >= 0) && (y1 < HH);
        const bool vx0 = (x0 >= 0) && (x0 < WW);
        const bool vx1 = (x1 >= 0) && (x1 < WW);

        // bilinear weights, mask folded in, zeroed when tap invalid
        const float w00 = (1.f - wy) * (1.f - wx) * ((vy0 && vx0) ? mm : 0.f);
        const float w01 = (1.f - wy) * wx         * ((vy0 && vx1) ? mm : 0.f);
        const float w10 = wy         * (1.f - wx) * ((vy1 && vx0) ? mm : 0.f);
        const float w11 = wy         * wx         * ((vy1 && vx1) ? mm : 0.f);

        const int yc0 = min(max(y0, 0), HH - 1), yc1 = min(max(y1, 0), HH - 1);
        const int xc0 = min(max(x0, 0), WW - 1), xc1 = min(max(x1, 0), WW - 1);
        const int a00 = yc0 * WW + xc0, a01 = yc0 * WW + xc1;
        const int a10 = yc1 * WW + xc0, a11 = yc1 * WW + xc1;

        const float* inb = input + (size_t)b * CC * HH * WW;
        #pragma unroll 4
        for (int c = 0; c < CC; ++c) {
            const float* ic = inb + c * (HH * WW);
            const float s = w00 * ic[a00] + w01 * ic[a01]
                          + w10 * ic[a10] + w11 * ic[a11];
            M[(c * KK + k) * MSTRIDE + pix] = s;
        }
    }
    __syncthreads();

    // ---------------- Phase B: D[64x64] = W[64x576] * M[576x64] ----------------
    // wave -> (Cout tile, 2 pixel tiles); 8 waves cover 4x4 tiles of 16x16.
    const int wave = tid >> 5;
    const int lane = tid & 31;
    const int lh   = lane >> 4;         // half-wave: selects K pair {0,1} vs {2,3}
    const int ll   = lane & 15;

    const int o_base = (wave & 3) * 16;
    const int pt     = wave >> 2;       // 0..1
    const int pixb0  = pt * 16;
    const int pixb1  = (pt + 2) * 16;

    v8f acc0 = {};
    v8f acc1 = {};

    // A-fragment source: 16x4 fp32 tile of weight. Lane ll = row o_base+ll,
    // VGPR0/1 = K = 2*lh + {0,1}. Weight row is contiguous in ck -> float2 load.
    const float* wrow = weight + (size_t)(o_base + ll) * CK + 2 * lh;

    #pragma unroll 4
    for (int kk = 0; kk < CK; kk += 4) {
        v2f a;
        a.x = wrow[kk];
        a.y = wrow[kk + 1];

        const int r = (kk + 2 * lh) * MSTRIDE;  // B rows K = 2*lh, 2*lh+1
        v2f b0, b1;
        b0.x = M[r + pixb0 + ll];
        b0.y = M[r + MSTRIDE + pixb0 + ll];
        b1.x = M[r + pixb1 + ll];
        b1.y = M[r + MSTRIDE + pixb1 + ll];

        acc0 = __builtin_amdgcn_wmma_f32_16x16x4_f32(
            false, a, false, b0, (short)0, acc0, false, false);
        acc1 = __builtin_amdgcn_wmma_f32_16x16x4_f32(
            false, a, false, b1, (short)0, acc1, false, false);
    }

    // D layout: VGPR r -> M = r (lanes 0-15) / M = 8+r (lanes 16-31), N = ll.
    float* ob = out + (size_t)b * COUT * HH * WW + pixg0;
    #pragma unroll
    for (int r2 = 0; r2 < 8; ++r2) {
        const int o0 = o_base + r2 + 8 * lh;
        const float bz = bias[o0];
        ob[(size_t)o0 * HH * WW + pixb0 + ll] = acc0[r2] + bz;
        ob[(size_t)o0 * HH * WW + pixb1 + ll] = acc1[r2] + bz;
    }
}

extern "C" void kernel_launch(void* const* d_in, const int* in_sizes, int n_in,
                              void* d_out, int out_size, void* d_ws, size_t ws_size,
                              hipStream_t stream) {
    const float* input  = (const float*)d_in[0];
    const float* offset = (const float*)d_in[1];
    const float* mask   = (const float*)d_in[2];
    const float* weight = (const float*)d_in[3];
    const float* bias   = (const float*)d_in[4];
    float* out = (float*)d_out;

    const size_t ldsBytes = (size_t)CK * MSTRIDE * sizeof(float); // 162 KB of 320 KB
    dim3 grid(16 * 64);   // B * (H rows), one 64-pixel row tile per block
    dim3 block(256);      // 8 waves (wave32)
    dcnv2_wmma_kernel<<<grid, block, ldsBytes, stream>>>(
        input, offset, mask, weight, bias, out);
}